// SparseSelfAttention_35493609734204
// MI455X (gfx1250) — compile-verified
//
#include <hip/hip_runtime.h>
#include <math.h>

// Problem constants (fixed by the reference harness)
#define B_  4
#define T_  1024
#define D_  1024
#define H_  8
#define DH_ 128
#define EPS_ 1e-10f
#define NEG_ -1e30f

// LDS tile pitches (floats)
#define APITCH 20    // A tile: 64 rows x 16 k, pitch 20 (16B-aligned rows, bank-conflict free)
#define BPITCH 260   // B tile: 8 pair-rows x (128 cols x 2), pitch 260

typedef __attribute__((ext_vector_type(2))) float v2f;
typedef __attribute__((ext_vector_type(4))) float v4f;
typedef __attribute__((ext_vector_type(8))) float v8f;
typedef __attribute__((ext_vector_type(2))) int   i2;
typedef __attribute__((ext_vector_type(4))) int   i4;

// Async-LDS builtin parameter types (confirmed by hipcc diagnostics):
//   b128: (int4 AS1*, int4 AS3*, imm offset, imm cpol)
//   b64:  (int2 AS1*, int2 AS3*, imm offset, imm cpol)
typedef __attribute__((address_space(1))) i4 g_i4;
typedef __attribute__((address_space(3))) i4 l_i4;
typedef __attribute__((address_space(1))) i2 g_i2;
typedef __attribute__((address_space(3))) i2 l_i2;

// D = A(16x4 f32) x B(4x16 f32) + C(16x16 f32), full-EXEC wave32
#define WMMA_F32(a, b, c) \
  __builtin_amdgcn_wmma_f32_16x16x4_f32(false, (a), false, (b), (short)0, (c), false, false)

#if defined(__gfx1250__) && \
    __has_builtin(__builtin_amdgcn_global_load_async_to_lds_b128) && \
    __has_builtin(__builtin_amdgcn_global_load_async_to_lds_b64)
#define USE_ASYNC_LDS 1
#else
#define USE_ASYNC_LDS 0
#endif

// s_wait_asynccnt needs a constant immediate -> template parameter
template <int N>
__device__ inline void wait_async() {
#if __has_builtin(__builtin_amdgcn_s_wait_asynccnt)
  __builtin_amdgcn_s_wait_asynccnt(N);
#else
  asm volatile("s_wait_asynccnt %0" :: "i"(N) : "memory");
#endif
}

// ---- WMMA fragment loaders from LDS tiles ------------------------------
// A 16x4 layout: VGPR0 = K {0 | 2}, VGPR1 = K {1 | 3}; lanes 0-15 vs 16-31
// kb is always even -> single aligned ds_load_b64
__device__ inline v2f afrag(const float* As, int m0, int kk, int lane) {
  int m  = m0 + (lane & 15);
  int kb = kk + ((lane >> 4) << 1);
  return *(const v2f*)&As[m * APITCH + kb];
}
// B stored K-pair interleaved: Bs[p*BPITCH + 2n + {0,1}] = B[2p|2p+1][n]
__device__ inline v2f bfrag(const float* Bs, int kk, int n0, int lane) {
  int n  = n0 + (lane & 15);
  int kb = kk + ((lane >> 4) << 1);
  return *(const v2f*)&Bs[(kb >> 1) * BPITCH + 2 * n];
}

// ---- x = where(pad, 0, inputs) -----------------------------------------
__global__ __launch_bounds__(256) void build_x_kernel(const float* __restrict__ in,
                                                      const int* __restrict__ lengths,
                                                      float* __restrict__ x) {
  size_t i = (size_t)blockIdx.x * 256 + threadIdx.x;   // over B*T*D
  int row = (int)(i / D_);                              // b*T + t
  int b = row >> 10, t = row & (T_ - 1);
  x[i] = (t < lengths[b]) ? in[i] : 0.0f;
}

// ---- C[M,N] = A[M,K] * W[N,K]^T  (row-major, WMMA f32) -----------------
// block: 256 thr = 8 waves; tile 64(M) x 128(N); wave tile 16 x 64
// Double-buffered async global->LDS pipeline: stage k+1's DMA is in flight
// while stage k's WMMAs run (in-order ASYNCcnt completion => wait(batch)).
__global__ __launch_bounds__(256) void gemm_xwT_kernel(const float* __restrict__ A,
                                                       const float* __restrict__ W,
                                                       float* __restrict__ C,
                                                       int M, int N, int K) {
  __shared__ float As[2][64 * APITCH];
  __shared__ float Bs[2][8 * BPITCH];
  int tid = threadIdx.x, lane = tid & 31, wave = tid >> 5;
  int m0 = blockIdx.y * 64, n0 = blockIdx.x * 128;
  int wm = (wave & 3) * 16, wn = (wave >> 2) * 64;
  v8f acc[4] = {};
#if USE_ASYNC_LDS
  auto issue = [&](int k0, int buf) {  // 5 async instrs per wave per stage
    int m = tid >> 2, kc = (tid & 3) * 4;
    __builtin_amdgcn_global_load_async_to_lds_b128(
        (g_i4*)&A[(size_t)(m0 + m) * K + k0 + kc],
        (l_i4*)&As[buf][m * APITCH + kc], 0, 0);
#pragma unroll
    for (int j = 0; j < 4; ++j) {
      int i = tid + j * 256;       // 1024 K-pair units
      int n = i >> 3, pk = i & 7;
      __builtin_amdgcn_global_load_async_to_lds_b64(
          (g_i2*)&W[(size_t)(n0 + n) * K + k0 + 2 * pk],
          (l_i2*)&Bs[buf][pk * BPITCH + 2 * n], 0, 0);
    }
  };
  issue(0, 0);
  for (int k0 = 0; k0 < K; k0 += 16) {
    int buf = (k0 >> 4) & 1;
    __syncthreads();                  // buf^1 fully consumed by all waves
    if (k0 + 16 < K) {
      issue(k0 + 16, buf ^ 1);        // prefetch next stage
      wait_async<5>();                // <=5 outstanding => current stage landed
    } else {
      wait_async<0>();
    }
    __syncthreads();                  // all waves' DMA for buf complete
    const float* Asb = As[buf];
    const float* Bsb = Bs[buf];
#pragma unroll
    for (int kk = 0; kk < 16; kk += 4) {
      v2f a = afrag(Asb, wm, kk, lane);
#pragma unroll
      for (int nt = 0; nt < 4; ++nt) {
        v2f b = bfrag(Bsb, kk, wn + nt * 16, lane);
        acc[nt] = WMMA_F32(a, b, acc[nt]);
      }
    }
  }
#else
  for (int k0 = 0; k0 < K; k0 += 16) {
    __syncthreads();
    {  // A: vectorized 16B staging
      int m = tid >> 2, kc = (tid & 3) * 4;
      *(v4f*)&As[0][m * APITCH + kc] = *(const v4f*)&A[(size_t)(m0 + m) * K + k0 + kc];
    }
#pragma unroll
    for (int j = 0; j < 2; ++j) {  // B: 16B load, two 8B pair stores
      int i = tid + j * 256;
      int n = i >> 2, kc = (i & 3) * 4;
      v4f w = *(const v4f*)&W[(size_t)(n0 + n) * K + k0 + kc];
      int p = kc >> 1;
      v2f lo; lo.x = w.x; lo.y = w.y;
      v2f hi; hi.x = w.z; hi.y = w.w;
      *(v2f*)&Bs[0][p * BPITCH + 2 * n] = lo;
      *(v2f*)&Bs[0][(p + 1) * BPITCH + 2 * n] = hi;
    }
    __syncthreads();
#pragma unroll
    for (int kk = 0; kk < 16; kk += 4) {
      v2f a = afrag(As[0], wm, kk, lane);
#pragma unroll
      for (int nt = 0; nt < 4; ++nt) {
        v2f b = bfrag(Bs[0], kk, wn + nt * 16, lane);
        acc[nt] = WMMA_F32(a, b, acc[nt]);
      }
    }
  }
#endif
  int rbase = m0 + wm + ((lane >> 4) << 3);
  int cbase = n0 + wn + (lane & 15);
#pragma unroll
  for (int nt = 0; nt < 4; ++nt)
#pragma unroll
    for (int r = 0; r < 8; ++r)
      C[(size_t)(rbase + r) * N + cbase + nt * 16] = acc[nt][r];
}

// ---- fused scores + edge + gate + mask ---------------------------------
// per (b,h): S = Q K^T, E = Eq Ek^T over dh=128; epilogue writes
// edges_prob = sigmoid(E) (nontemporal) and masked raw scores.
__global__ __launch_bounds__(256) void score_kernel(const float* __restrict__ q,
                                                    const float* __restrict__ k,
                                                    const float* __restrict__ eq,
                                                    const float* __restrict__ ek,
                                                    const float* __restrict__ noise,
                                                    const int* __restrict__ lengths,
                                                    float* __restrict__ attn,
                                                    float* __restrict__ edges) {
  __shared__ float Aq[2][64 * APITCH], Ae[2][64 * APITCH];
  __shared__ float Bk[2][8 * BPITCH], Be[2][8 * BPITCH];
  int tid = threadIdx.x, lane = tid & 31, wave = tid >> 5;
  int bh = blockIdx.z, b = bh >> 3, h = bh & 7;
  int qt0 = blockIdx.y * 64, kt0 = blockIdx.x * 128;
  int len = lengths[b];
  size_t base = (size_t)b * T_ * D_ + (size_t)h * DH_;
  int wm = (wave & 3) * 16, wn = (wave >> 2) * 64;
  v8f aS[4] = {}, aE[4] = {};
#if USE_ASYNC_LDS
  auto issue = [&](int k0, int buf) {  // 10 async instrs per wave per stage
    int m = tid >> 2, kc = (tid & 3) * 4;
    size_t aoff = base + (size_t)(qt0 + m) * D_ + k0 + kc;
    __builtin_amdgcn_global_load_async_to_lds_b128(
        (g_i4*)&q[aoff],  (l_i4*)&Aq[buf][m * APITCH + kc], 0, 0);
    __builtin_amdgcn_global_load_async_to_lds_b128(
        (g_i4*)&eq[aoff], (l_i4*)&Ae[buf][m * APITCH + kc], 0, 0);
#pragma unroll
    for (int j = 0; j < 4; ++j) {
      int i = tid + j * 256;
      int n = i >> 3, pk = i & 7;
      size_t boff = base + (size_t)(kt0 + n) * D_ + k0 + 2 * pk;
      int lds = pk * BPITCH + 2 * n;
      __builtin_amdgcn_global_load_async_to_lds_b64(
          (g_i2*)&k[boff],  (l_i2*)&Bk[buf][lds], 0, 0);
      __builtin_amdgcn_global_load_async_to_lds_b64(
          (g_i2*)&ek[boff], (l_i2*)&Be[buf][lds], 0, 0);
    }
  };
  issue(0, 0);
#endif
  for (int k0 = 0; k0 < DH_; k0 += 16) {
    int buf = (k0 >> 4) & 1;
    __syncthreads();
#if USE_ASYNC_LDS
    if (k0 + 16 < DH_) {
      issue(k0 + 16, buf ^ 1);
      wait_async<10>();
    } else {
      wait_async<0>();
    }
#else
    {  // A tiles (q, eq)
      int m = tid >> 2, kc = (tid & 3) * 4;
      size_t off = base + (size_t)(qt0 + m) * D_ + k0 + kc;
      *(v4f*)&Aq[buf][m * APITCH + kc] = *(const v4f*)&q[off];
      *(v4f*)&Ae[buf][m * APITCH + kc] = *(const v4f*)&eq[off];
    }
#pragma unroll
    for (int j = 0; j < 2; ++j) {  // B tiles (k, ek) -> pair-interleaved
      int i = tid + j * 256;
      int n = i >> 2, kc = (i & 3) * 4;
      size_t off = base + (size_t)(kt0 + n) * D_ + k0 + kc;
      int p = kc >> 1;
      v4f w1 = *(const v4f*)&k[off];
      v2f lo; lo.x = w1.x; lo.y = w1.y;
      v2f hi; hi.x = w1.z; hi.y = w1.w;
      *(v2f*)&Bk[buf][p * BPITCH + 2 * n] = lo;
      *(v2f*)&Bk[buf][(p + 1) * BPITCH + 2 * n] = hi;
      v4f w2 = *(const v4f*)&ek[off];
      lo.x = w2.x; lo.y = w2.y;
      hi.x = w2.z; hi.y = w2.w;
      *(v2f*)&Be[buf][p * BPITCH + 2 * n] = lo;
      *(v2f*)&Be[buf][(p + 1) * BPITCH + 2 * n] = hi;
    }
#endif
    __syncthreads();
#pragma unroll
    for (int kk = 0; kk < 16; kk += 4) {
      v2f a1 = afrag(Aq[buf], wm, kk, lane);
      v2f a2 = afrag(Ae[buf], wm, kk, lane);
#pragma unroll
      for (int nt = 0; nt < 4; ++nt) {
        v2f b1 = bfrag(Bk[buf], kk, wn + nt * 16, lane);
        v2f b2 = bfrag(Be[buf], kk, wn + nt * 16, lane);
        aS[nt] = WMMA_F32(a1, b1, aS[nt]);
        aE[nt] = WMMA_F32(a2, b2, aE[nt]);
      }
    }
  }
  const float inv_sqrt_d = 0.03125f;  // 1/sqrt(1024)
  int qbase = qt0 + wm + ((lane >> 4) << 3);
  int kbase = kt0 + wn + (lane & 15);
  size_t rowbase = (size_t)bh * T_ * T_;
#pragma unroll
  for (int nt = 0; nt < 4; ++nt) {
    int key = kbase + nt * 16;
    bool keyok = key < len;
#pragma unroll
    for (int r = 0; r < 8; ++r) {
      int qrow = qbase + r;
      size_t idx = rowbase + (size_t)qrow * T_ + key;
      float e = aE[nt][r];
      float p = 1.0f / (1.0f + __expf(-e));
      __builtin_nontemporal_store(p, &edges[idx]);        // write-once output
      float u = __builtin_nontemporal_load(&noise[idx]);  // read-once input
      float logistic = __logf(u + EPS_) - __logf(1.0f - u + EPS_);
      float z = __logf(p + EPS_) - __logf(1.0f - p + EPS_) + logistic;
      bool gate = z > 0.0f;  // sigmoid(z) > 0.5  <=>  z > 0
      bool valid = gate && keyok;
      attn[idx] = valid ? aS[nt][r] * inv_sqrt_d : NEG_;  // keep RT: re-read twice
    }
  }
}

// ---- in-place row softmax over attn; fully-masked rows -> 0 ------------
__global__ __launch_bounds__(256) void softmax_kernel(float* __restrict__ attn) {
  __shared__ float red[256];
  float* p = attn + (size_t)blockIdx.x * T_;
  int tid = threadIdx.x;
  float v[4];
  float m = NEG_;
#pragma unroll
  for (int j = 0; j < 4; ++j) { v[j] = p[tid + j * 256]; m = fmaxf(m, v[j]); }
  red[tid] = m; __syncthreads();
  for (int s = 128; s > 0; s >>= 1) { if (tid < s) red[tid] = fmaxf(red[tid], red[tid + s]); __syncthreads(); }
  float rm = red[0]; __syncthreads();
  if (rm <= -1e29f) {  // no valid key in this row
#pragma unroll
    for (int j = 0; j < 4; ++j) p[tid + j * 256] = 0.0f;
    return;
  }
  float s = 0.0f, e[4];
#pragma unroll
  for (int j = 0; j < 4; ++j) { e[j] = __expf(v[j] - rm); s += e[j]; }
  red[tid] = s; __syncthreads();
  for (int st = 128; st > 0; st >>= 1) { if (tid < st) red[tid] += red[tid + st]; __syncthreads(); }
  float inv = 1.0f / red[0];
#pragma unroll
  for (int j = 0; j < 4; ++j) p[tid + j * 256] = e[j] * inv;
}

// ---- ctx = relu(attn @ v) per head -------------------------------------
__global__ __launch_bounds__(256) void ctx_kernel(const float* __restrict__ attn,
                                                  const float* __restrict__ v,
                                                  float* __restrict__ ctx) {
  __shared__ float As[64 * APITCH];
  __shared__ float Bs[8 * BPITCH];
  int tid = threadIdx.x, lane = tid & 31, wave = tid >> 5;
  int bh = blockIdx.z, b = bh >> 3, h = bh & 7;
  int m0 = blockIdx.y * 64;
  size_t abase = (size_t)bh * T_ * T_;
  size_t vbase = (size_t)b * T_ * D_ + (size_t)h * DH_;
  int wm = (wave & 3) * 16, wn = (wave >> 2) * 64;  // wn in {0,64}, N = dh = 128
  v8f acc[4] = {};
  for (int k0 = 0; k0 < T_; k0 += 16) {
    __syncthreads();
    {  // A = attn tile
      int m = tid >> 2, kc = (tid & 3) * 4;
      *(v4f*)&As[m * APITCH + kc] =
          *(const v4f*)&attn[abase + (size_t)(m0 + m) * T_ + k0 + kc];
    }
    {  // B = v tile (row-major in memory) -> pair-interleave two k rows
      int pr = tid >> 5;           // pair-row 0..7
      int nq = (tid & 31) * 4;     // col quad
      const float* r0 = &v[vbase + (size_t)(k0 + 2 * pr) * D_ + nq];
      v4f a0 = *(const v4f*)r0;
      v4f a1 = *(const v4f*)(r0 + D_);
      v4f lo; lo.x = a0.x; lo.y = a1.x; lo.z = a0.y; lo.w = a1.y;
      v4f hi; hi.x = a0.z; hi.y = a1.z; hi.z = a0.w; hi.w = a1.w;
      *(v4f*)&Bs[pr * BPITCH + 2 * nq] = lo;
      *(v4f*)&Bs[pr * BPITCH + 2 * nq + 4] = hi;
    }
    __syncthreads();
#pragma unroll
    for (int kk = 0; kk < 16; kk += 4) {
      v2f a = afrag(As, wm, kk, lane);
#pragma unroll
      for (int nt = 0; nt < 4; ++nt) {
        v2f bf = bfrag(Bs, kk, wn + nt * 16, lane);
        acc[nt] = WMMA_F32(a, bf, acc[nt]);
      }
    }
  }
  int rbase = m0 + wm + ((lane >> 4) << 3);
  int cbase = wn + (lane & 15);
#pragma unroll
  for (int nt = 0; nt < 4; ++nt)
#pragma unroll
    for (int r = 0; r < 8; ++r)
      ctx[vbase + (size_t)(rbase + r) * D_ + cbase + nt * 16] = fmaxf(acc[nt][r], 0.0f);
}

// ---- h1 = layernorm(ctx + x) -------------------------------------------
__global__ __launch_bounds__(256) void ln1_kernel(const float* __restrict__ ctx,
                                                  const float* __restrict__ x,
                                                  const float* __restrict__ g,
                                                  const float* __restrict__ beta,
                                                  float* __restrict__ h1) {
  __shared__ float red[256];
  size_t row = blockIdx.x;
  int tid = threadIdx.x;
  float y[4];
  float s = 0.0f;
#pragma unroll
  for (int j = 0; j < 4; ++j) {
    size_t idx = row * D_ + tid + j * 256;
    y[j] = ctx[idx] + x[idx];
    s += y[j];
  }
  red[tid] = s; __syncthreads();
  for (int st = 128; st > 0; st >>= 1) { if (tid < st) red[tid] += red[tid + st]; __syncthreads(); }
  float mu = red[0] * (1.0f / D_); __syncthreads();
  float vs = 0.0f;
#pragma unroll
  for (int j = 0; j < 4; ++j) { float d = y[j] - mu; vs += d * d; }
  red[tid] = vs; __syncthreads();
  for (int st = 128; st > 0; st >>= 1) { if (tid < st) red[tid] += red[tid + st]; __syncthreads(); }
  float inv = rsqrtf(red[0] * (1.0f / D_) + 1e-5f);
#pragma unroll
  for (int j = 0; j < 4; ++j) {
    int d2 = tid + j * 256;
    h1[row * D_ + d2] = (y[j] - mu) * inv * g[d2] + beta[d2];
  }
}

// ---- out = layernorm(relu(fc + bfc) + h1), zero padded rows ------------
__global__ __launch_bounds__(256) void ln2_kernel(const float* __restrict__ fc,
                                                  const float* __restrict__ h1,
                                                  const float* __restrict__ bfc,
                                                  const float* __restrict__ g,
                                                  const float* __restrict__ beta,
                                                  const int* __restrict__ lengths,
                                                  float* __restrict__ out) {
  __shared__ float red[256];
  size_t row = blockIdx.x;
  int b = (int)(row >> 10), t = (int)(row & (T_ - 1));
  int tid = threadIdx.x;
  if (t >= lengths[b]) {  // uniform branch across block
#pragma unroll
    for (int j = 0; j < 4; ++j) out[row * D_ + tid + j * 256] = 0.0f;
    return;
  }
  float y[4];
  float s = 0.0f;
#pragma unroll
  for (int j = 0; j < 4; ++j) {
    int d2 = tid + j * 256;
    size_t idx = row * D_ + d2;
    y[j] = fmaxf(fc[idx] + bfc[d2], 0.0f) + h1[idx];
    s += y[j];
  }
  red[tid] = s; __syncthreads();
  for (int st = 128; st > 0; st >>= 1) { if (tid < st) red[tid] += red[tid + st]; __syncthreads(); }
  float mu = red[0] * (1.0f / D_); __syncthreads();
  float vs = 0.0f;
#pragma unroll
  for (int j = 0; j < 4; ++j) { float d = y[j] - mu; vs += d * d; }
  red[tid] = vs; __syncthreads();
  for (int st = 128; st > 0; st >>= 1) { if (tid < st) red[tid] += red[tid + st]; __syncthreads(); }
  float inv = rsqrtf(red[0] * (1.0f / D_) + 1e-5f);
#pragma unroll
  for (int j = 0; j < 4; ++j) {
    int d2 = tid + j * 256;
    out[row * D_ + d2] = (y[j] - mu) * inv * g[d2] + beta[d2];
  }
}

extern "C" void kernel_launch(void* const* d_in, const int* in_sizes, int n_in,
                              void* d_out, int out_size, void* d_ws, size_t ws_size,
                              hipStream_t stream) {
  (void)in_sizes; (void)n_in; (void)out_size; (void)ws_size;
  const float* inputs  = (const float*)d_in[0];
  const int*   lengths = (const int*)d_in[1];
  const float* noise   = (const float*)d_in[2];
  const float* Wq  = (const float*)d_in[3];
  const float* Wk  = (const float*)d_in[4];
  const float* Wv  = (const float*)d_in[5];
  const float* Weq = (const float*)d_in[6];
  const float* Wek = (const float*)d_in[7];
  const float* Wfc = (const float*)d_in[8];
  const float* bfc = (const float*)d_in[9];
  const float* ln1_g = (const float*)d_in[10];
  const float* ln1_b = (const float*)d_in[11];
  const float* ln2_g = (const float*)d_in[12];
  const float* ln2_b = (const float*)d_in[13];

  const size_t NTD = (size_t)B_ * T_ * D_;       // 4,194,304
  const size_t NATT = (size_t)B_ * H_ * T_ * T_; // 33,554,432

  float* out   = (float*)d_out;
  float* attn  = out + NTD;
  float* edges = attn + NATT;

  float* ws  = (float*)d_ws;
  float* x   = ws + 0 * NTD;
  float* q   = ws + 1 * NTD;
  float* k   = ws + 2 * NTD;
  float* v   = ws + 3 * NTD;
  float* eq  = ws + 4 * NTD;
  float* ek  = ws + 5 * NTD;
  float* ctx = ws + 6 * NTD;
  float* h1  = ws + 7 * NTD;
  float* fct = q;  // q is dead after score; reuse for FC output

  // 1) masked input
  build_x_kernel<<<(int)(NTD / 256), 256, 0, stream>>>(inputs, lengths, x);

  // 2) five projections: h = x @ W^T
  dim3 gg(D_ / 128, (B_ * T_) / 64);
  gemm_xwT_kernel<<<gg, 256, 0, stream>>>(x, Wq,  q,  B_ * T_, D_, D_);
  gemm_xwT_kernel<<<gg, 256, 0, stream>>>(x, Wk,  k,  B_ * T_, D_, D_);
  gemm_xwT_kernel<<<gg, 256, 0, stream>>>(x, Wv,  v,  B_ * T_, D_, D_);
  gemm_xwT_kernel<<<gg, 256, 0, stream>>>(x, Weq, eq, B_ * T_, D_, D_);
  gemm_xwT_kernel<<<gg, 256, 0, stream>>>(x, Wek, ek, B_ * T_, D_, D_);

  // 3) scores + edges + gate + mask (writes edges_prob and masked raw scores)
  dim3 gs(T_ / 128, T_ / 64, B_ * H_);
  score_kernel<<<gs, 256, 0, stream>>>(q, k, eq, ek, noise, lengths, attn, edges);

  // 4) row softmax in place
  softmax_kernel<<<B_ * H_ * T_, 256, 0, stream>>>(attn);

  // 5) ctx = relu(attn @ v)
  dim3 gc(1, T_ / 64, B_ * H_);
  ctx_kernel<<<gc, 256, 0, stream>>>(attn, v, ctx);

  // 6) h1 = LN(ctx + x)
  ln1_kernel<<<B_ * T_, 256, 0, stream>>>(ctx, x, ln1_g, ln1_b, h1);

  // 7) FC GEMM: fct = h1 @ Wfc^T
  gemm_xwT_kernel<<<gg, 256, 0, stream>>>(h1, Wfc, fct, B_ * T_, D_, D_);

  // 8) out = LN(relu(fct + bfc) + h1), zero padded rows
  ln2_kernel<<<B_ * T_, 256, 0, stream>>>(fct, h1, bfc, ln2_g, ln2_b, lengths, out);
}